// MotionOutputLayers_79448305041769
// MI455X (gfx1250) — compile-verified
//
#include <hip/hip_runtime.h>
#include <hip/hip_bf16.h>
#include <stdint.h>

#define N_CAND 6144          // R*K
#define R_ROWS 2048
#define K_CLS 3
#define SORT_N 8192
#define TOPK 100
#define SCORE_THRESH 0.05f
#define NMS_THRESH 0.5f
#define TILE16 (N_CAND / 16)          // 384 tiles per dimension
#define MASK_U16_PER_ROW TILE16       // 384 u16 per mask row (768 B)
#define MASK_U32_PER_ROW (N_CAND/32)  // 192 u32 per mask row
#define QDEPTH 8                      // async LDS ring depth (power of two)

typedef __attribute__((ext_vector_type(2))) float v2f;
typedef __attribute__((ext_vector_type(8))) float v8f;

// ---------------- Kernel A: clip boxes, extract fg scores, global max coord --
__global__ void prep_kernel(const float* __restrict__ boxes,
                            const float* __restrict__ scores,
                            const int* __restrict__ image_h,
                            const int* __restrict__ image_w,
                            float* __restrict__ bf, float* __restrict__ s,
                            unsigned* __restrict__ maxbits)
{
    int i = blockIdx.x * blockDim.x + threadIdx.x;
    if (i >= N_CAND) return;
    int r = i / K_CLS, c = i % K_CLS;
    float W = (float)image_w[0], H = (float)image_h[0];
    const float* bp = boxes + (size_t)r * (K_CLS * 4) + c * 4;
    float x1 = fminf(fmaxf(bp[0], 0.f), W);
    float y1 = fminf(fmaxf(bp[1], 0.f), H);
    float x2 = fminf(fmaxf(bp[2], 0.f), W);
    float y2 = fminf(fmaxf(bp[3], 0.f), H);
    ((float4*)bf)[i] = make_float4(x1, y1, x2, y2);
    s[i] = scores[(size_t)r * (K_CLS + 1) + c];
    // clipped coords are all >= 0 -> IEEE bits are monotone under uint compare
    float m = fmaxf(fmaxf(x1, y1), fmaxf(x2, y2));
    atomicMax(maxbits, __float_as_uint(m));
}

// ---------------- Kernel B: single-WG bitonic argsort (desc by masked score) -
__global__ __launch_bounds__(1024) void sort_kernel(
    const float* __restrict__ s_arr, const float* __restrict__ bf,
    const unsigned* __restrict__ maxbits,
    float* __restrict__ ss, float* __restrict__ bs, float* __restrict__ ob,
    int* __restrict__ cls_s, int* __restrict__ pr_s)
{
    __shared__ float key[SORT_N];   // ascending sort on -score (invalid -> +inf)
    __shared__ int   val[SORT_N];
    int t = threadIdx.x;
    for (int i = t; i < SORT_N; i += 1024) {
        if (i < N_CAND) {
            float sv = s_arr[i];
            key[i] = (sv > SCORE_THRESH) ? -sv : __builtin_inff();
        } else {
            key[i] = __builtin_inff();
        }
        val[i] = i;
    }
    __syncthreads();
    for (int k = 2; k <= SORT_N; k <<= 1) {
        for (int j = k >> 1; j > 0; j >>= 1) {
            for (int i = t; i < SORT_N; i += 1024) {
                int ixj = i ^ j;
                if (ixj > i) {
                    bool dirUp = ((i & k) == 0);
                    float ki = key[i], kj = key[ixj];
                    if ((ki > kj) == dirUp) {
                        int vi = val[i], vj = val[ixj];
                        key[i] = kj; key[ixj] = ki;
                        val[i] = vj; val[ixj] = vi;
                    }
                }
            }
            __syncthreads();
        }
    }
    float maxc = __uint_as_float(*maxbits);
    for (int i = t; i < N_CAND; i += 1024) {
        int o = val[i];
        if (o < N_CAND) {
            int c = o % K_CLS, p = o / K_CLS;
            float4 b = ((const float4*)bf)[o];
            ((float4*)bs)[i] = b;
            float off = (float)c * (maxc + 1.0f);
            ((float4*)ob)[i] = make_float4(b.x + off, b.y + off, b.z + off, b.w + off);
            ss[i] = s_arr[o];
            cls_s[i] = c;
            pr_s[i] = p;
        } else {  // padding slots interleaved among the -inf block
            ((float4*)bs)[i] = make_float4(0.f, 0.f, 0.f, 0.f);
            ((float4*)ob)[i] = make_float4(0.f, 0.f, 0.f, 0.f);
            ss[i] = -1.0f;  // <= thresh -> pre-suppressed
            cls_s[i] = 0;
            pr_s[i] = 0;
        }
    }
}

// ---------------- Kernel C: pairwise IoU>thresh bitmask via WMMA f32 16x16x4 -
// Each wave computes one 16x16 tile. Pairwise sums/diffs are rank-2 outer
// products: D[m][n] = p_m*1 + 1*(+-q_n) on the matrix pipe; nonlinear
// min/max/abs/compare finishing runs on the 8 accumulator VGPRs.
__global__ __launch_bounds__(256) void iou_mask_kernel(
    const float* __restrict__ ob, uint16_t* __restrict__ mask)
{
    int gid  = blockIdx.x * blockDim.x + threadIdx.x;
    int wave = gid >> 5;
    int lane = threadIdx.x & 31;
    int tileR = wave / TILE16;
    int tileC = wave % TILE16;
    bool isRow = lane < 16;
    int idx = isRow ? (tileR * 16 + lane) : (tileC * 16 + (lane - 16));

    float4 b = ((const float4*)ob)[idx];
    float x1 = b.x, y1 = b.y, x2 = b.z, y2 = b.w;
    float w = x2 - x1, h = y2 - y1;
    float area = fmaxf(w, 0.f) * fmaxf(h, 0.f);

    // A (16x4 f32): lanes 0-15 hold K0/K1 = (p, 1); lanes 16-31 hold K2/K3 = 0.
    // B (4x16 f32): VGPR0 lanes 0-15 = K0 row (ones), lanes 16-31 = K1 row (q).
    auto pairwise = [&](float p, float q) -> v8f {
        v2f A, B;
        A[0] = isRow ? p : 0.0f;
        A[1] = isRow ? 1.0f : 0.0f;
        B[0] = isRow ? 1.0f : q;
        B[1] = 0.0f;
        v8f c = {};
        return __builtin_amdgcn_wmma_f32_16x16x4_f32(
            false, A, false, B, (short)0, c, false, false);
    };

    v8f Sw  = pairwise(w,    w);     // w_i + w_j
    v8f Dx1 = pairwise(x1,  -x1);    // x1_i - x1_j
    v8f Dx2 = pairwise(x2,  -x2);    // x2_i - x2_j
    v8f Sh  = pairwise(h,    h);     // h_i + h_j
    v8f Dy1 = pairwise(y1,  -y1);
    v8f Dy2 = pairwise(y2,  -y2);
    v8f Sa  = pairwise(area, area);  // area_i + area_j

    // C/D layout: VGPR r -> rows (tileR*16 + r) [lanes 0-15] and (+8) [16-31].
    int rowBase = tileR * 16;
#pragma unroll
    for (int rl = 0; rl < 8; rl++) {
        // inter_w = max(0, min(x2i,x2j) - max(x1i,x1j))
        //         = max(0, 0.5*((wi+wj) - |x2i-x2j| - |x1i-x1j|))
        float iw = fmaxf(0.f, 0.5f * (Sw[rl] - fabsf(Dx1[rl]) - fabsf(Dx2[rl])));
        float ih = fmaxf(0.f, 0.5f * (Sh[rl] - fabsf(Dy1[rl]) - fabsf(Dy2[rl])));
        float inter = iw * ih;
        float uni = fmaxf(Sa[rl] - inter, 1e-9f);
        unsigned bal = __builtin_amdgcn_ballot_w32(inter > NMS_THRESH * uni);
        if (lane == 0) {
            mask[(size_t)(rowBase + rl)     * MASK_U16_PER_ROW + tileC] = (uint16_t)(bal & 0xFFFFu);
            mask[(size_t)(rowBase + rl + 8) * MASK_U16_PER_ROW + tileC] = (uint16_t)(bal >> 16);
        }
    }
}

// ---------------- Kernel D: sequential NMS scan + top-100 gather -------------
// 192 threads: thread t owns suppression word t (32 candidates). Mask rows are
// streamed into an 8-deep LDS ring with global_load_async_to_lds_b32 so the
// serial loop never takes a demand global-latency hit; s_wait_asynccnt 7
// guarantees (in-order async completion) that the oldest row is resident.
__global__ __launch_bounds__(192) void nms_scan_kernel(
    const uint16_t* __restrict__ mask,
    const float* __restrict__ ss, const float* __restrict__ bs,
    const int* __restrict__ cls_s, const int* __restrict__ pr_s,
    const float* __restrict__ mtype, const float* __restrict__ morigin,
    const float* __restrict__ maxis, const float* __restrict__ mextr,
    float* __restrict__ out)
{
    __shared__ unsigned sup[MASK_U32_PER_ROW];
    __shared__ unsigned rowbuf[QDEPTH][MASK_U32_PER_ROW];  // 6 KB ring
    __shared__ int rank_s;
    int t = threadIdx.x;

    unsigned wv = 0;
    for (int bidx = 0; bidx < 32; bidx++) {
        int j = t * 32 + bidx;
        if (!(ss[j] > SCORE_THRESH)) wv |= (1u << bidx);  // pre-suppress invalid
    }
    sup[t] = wv;
    if (t == 0) rank_s = 0;

    // Prime the async ring: rows 0..QDEPTH-1, thread t fetches its own word,
    // so row data has no cross-wave LDS dependency (own-wave ASYNCcnt gates it).
    for (int d = 0; d < QDEPTH; d++) {
        unsigned ldsa = (unsigned)(unsigned long long)(&rowbuf[d][t]);
        unsigned goff = (unsigned)d * (MASK_U16_PER_ROW * 2) + (unsigned)t * 4u;
        asm volatile("global_load_async_to_lds_b32 %0, %1, %2"
                     :
                     : "v"(ldsa), "v"(goff), "s"(mask)
                     : "memory");
    }
    __syncthreads();

    for (int i = 0; i < N_CAND; i++) {
        bool keep = ((sup[i >> 5] >> (i & 31)) & 1u) == 0u;
        int rk = rank_s;
        __syncthreads();
        if (keep) {
            // row i is resident once <= QDEPTH-1 async loads remain outstanding
            asm volatile("s_wait_asynccnt 0x7" ::: "memory");
            unsigned m = rowbuf[i & (QDEPTH - 1)][t];
            int word = i >> 5;
            if (t < word)       m = 0u;
            else if (t == word) m &= ~((2u << (i & 31)) - 1u);  // keep only j > i
            sup[t] |= m;
            if (t == 0) {
                rank_s = rk + 1;
                if (rk < TOPK) {
                    float* det = out + rk * 26;
                    det[0] = bs[i * 4 + 0];
                    det[1] = bs[i * 4 + 1];
                    det[2] = bs[i * 4 + 2];
                    det[3] = bs[i * 4 + 3];
                    det[4] = ss[i];
                    int p = pr_s[i];
                    det[5]  = mtype[p * 3 + 0];  det[6]  = mtype[p * 3 + 1];  det[7]  = mtype[p * 3 + 2];
                    det[8]  = morigin[p * 3 + 0];det[9]  = morigin[p * 3 + 1];det[10] = morigin[p * 3 + 2];
                    det[11] = maxis[p * 3 + 0];  det[12] = maxis[p * 3 + 1];  det[13] = maxis[p * 3 + 2];
#pragma unroll
                    for (int q = 0; q < 12; q++) det[14 + q] = mextr[p * 12 + q];
                    out[TOPK * 26 + rk]            = (float)cls_s[i];  // cls_out
                    out[TOPK * 26 + TOPK + rk]     = (float)pr_s[i];   // idx_out
                    out[TOPK * 26 + 2 * TOPK + rk] = 1.0f;             // valid_out
                }
            }
        }
        // Refill the slot just consumed with row i+QDEPTH. DS reads and async
        // writes are unordered counters, so retire our LDS read first.
        if (i + QDEPTH < N_CAND) {
            asm volatile("s_wait_asynccnt 0x7" ::: "memory");
            asm volatile("s_wait_dscnt 0x0" ::: "memory");
            int slot = (i + QDEPTH) & (QDEPTH - 1);
            unsigned ldsa = (unsigned)(unsigned long long)(&rowbuf[slot][t]);
            unsigned goff = (unsigned)(i + QDEPTH) * (MASK_U16_PER_ROW * 2) + (unsigned)t * 4u;
            asm volatile("global_load_async_to_lds_b32 %0, %1, %2"
                         :
                         : "v"(ldsa), "v"(goff), "s"(mask)
                         : "memory");
        }
        __syncthreads();
    }
}

extern "C" void kernel_launch(void* const* d_in, const int* in_sizes, int n_in,
                              void* d_out, int out_size, void* d_ws, size_t ws_size,
                              hipStream_t stream)
{
    const float* boxes   = (const float*)d_in[0];
    const float* scores  = (const float*)d_in[1];
    const float* mtype   = (const float*)d_in[2];
    const float* morigin = (const float*)d_in[3];
    const float* maxis   = (const float*)d_in[4];
    const float* mextr   = (const float*)d_in[5];
    const int*   image_h = (const int*)d_in[6];
    const int*   image_w = (const int*)d_in[7];

    char* ws = (char*)d_ws;                      // all offsets 16B-aligned
    float*    bf      = (float*)(ws + 0);        // 6144*4 f32  clipped boxes
    float*    s       = (float*)(ws + 98304);    // 6144 f32    fg scores
    float*    ss      = (float*)(ws + 122880);   // 6144 f32    sorted scores
    float*    bs      = (float*)(ws + 147456);   // 6144*4 f32  sorted boxes
    float*    ob      = (float*)(ws + 245760);   // 6144*4 f32  class-offset boxes
    int*      cls_s   = (int*)(ws + 344064);     // 6144 i32
    int*      pr_s    = (int*)(ws + 368640);     // 6144 i32
    unsigned* maxbits = (unsigned*)(ws + 393216);
    uint16_t* mask    = (uint16_t*)(ws + 393280); // 6144*384 u16 = 4.72 MB

    hipMemsetAsync(maxbits, 0, sizeof(unsigned), stream);
    hipMemsetAsync(d_out, 0, (size_t)out_size * sizeof(float), stream);

    prep_kernel<<<(N_CAND + 255) / 256, 256, 0, stream>>>(
        boxes, scores, image_h, image_w, bf, s, maxbits);

    sort_kernel<<<1, 1024, 0, stream>>>(s, bf, maxbits, ss, bs, ob, cls_s, pr_s);

    int totalWaves = TILE16 * TILE16;  // 147456 tiles, 8 waves / 256-thread block
    iou_mask_kernel<<<totalWaves / 8, 256, 0, stream>>>(ob, mask);

    nms_scan_kernel<<<1, MASK_U32_PER_ROW, 0, stream>>>(
        mask, ss, bs, cls_s, pr_s, mtype, morigin, maxis, mextr, (float*)d_out);
}